// CachedGemmaAttention_3934190043185
// MI455X (gfx1250) — compile-verified
//
#include <hip/hip_runtime.h>
#include <hip/hip_bf16.h>

// ---------------------------------------------------------------------------
// Gemma cached attention decode step on MI455X (gfx1250, wave32).
// B=32, T=1, D=2048, S=4096, NQ=8, NKV=1, H=256, cache update @ idx.
// Strategy: WMMA f32 GEMMs for the projections (M=32 from batch dim),
// cache copy fused into the K/V streaming passes so each cache byte is
// read once and written once (~536 MB total => ~23us at 23.3 TB/s).
// ---------------------------------------------------------------------------

typedef float v2f __attribute__((ext_vector_type(2)));
typedef float v8f __attribute__((ext_vector_type(8)));

#define BATCH 32
#define DMODEL 2048
#define SEQ 4096
#define NQH 8
#define HDIM 256
#define PROJ_COLS 2560            // 8*256 (Q) + 256 (K) + 256 (V)
#define KV_BSTRIDE (2 * SEQ * HDIM)   // per-batch stride in cache
#define NEG_INF_F (-1.0e9f)

// gfx1250 async global->LDS path (ASYNCcnt) if the toolchain exposes it.
#if defined(__has_builtin)
#if __has_builtin(__builtin_amdgcn_global_load_async_to_lds_b32) && \
    __has_builtin(__builtin_amdgcn_s_wait_asynccnt)
#define HAVE_ASYNC_LDS 1
typedef __attribute__((address_space(1))) int async_gptr;
typedef __attribute__((address_space(3))) int async_lptr;
#endif
#endif

// ---------------------------------------------------------------------------
// K1: fused QKV projection.  P[32][2560] = X[32][2048] @ [Wq|Wk|Wv]
// One wave per 16x16 output tile.  2 row tiles * 160 col tiles = 320 waves.
// ---------------------------------------------------------------------------
__global__ void __launch_bounds__(256)
qkv_gemm_kernel(const float* __restrict__ x,
                const float* __restrict__ Wq,
                const float* __restrict__ Wk,
                const float* __restrict__ Wv,
                float* __restrict__ P)
{
    const int wave = (blockIdx.x * blockDim.x + threadIdx.x) >> 5;
    const int lane = threadIdx.x & 31;
    const int tile_m = wave / 160;          // 0..1
    const int tile_n = wave % 160;          // 0..159

    // Per-lane column pointer for the B fragment (stride 256 floats in k).
    const int col = tile_n * 16 + (lane & 15);
    const float* wcol;
    if (col < 2048) {
        wcol = Wq + (size_t)(col >> 8) * (DMODEL * HDIM) + (col & 255);
    } else if (col < 2304) {
        wcol = Wk + (col - 2048);
    } else {
        wcol = Wv + (col - 2304);
    }

    const int m = tile_m * 16 + (lane & 15);      // batch row for A fragment
    const int koff = (lane >> 4) << 1;            // lanes 16-31 hold K+2,K+3

    v8f acc = {};
    #pragma unroll 2
    for (int k = 0; k < DMODEL; k += 4) {
        const int kk = k + koff;
        v2f a, b;
        a.x = x[(size_t)m * DMODEL + kk];
        a.y = x[(size_t)m * DMODEL + kk + 1];
        b.x = wcol[(size_t)kk * HDIM];
        b.y = wcol[(size_t)(kk + 1) * HDIM];
        acc = __builtin_amdgcn_wmma_f32_16x16x4_f32(
            false, a, false, b, (short)0, acc, false, false);
    }

    // C/D layout: VGPR r -> row r (lanes 0-15), row r+8 (lanes 16-31).
    const int mrow_base = tile_m * 16 + ((lane >> 4) << 3);
    #pragma unroll
    for (int r = 0; r < 8; ++r) {
        P[(size_t)(mrow_base + r) * PROJ_COLS + col] = acc[r];
    }
}

// ---------------------------------------------------------------------------
// K2: RoPE on q and k_update, scale q by 1/sqrt(H), scatter new K/V row
// into the output cache at row idx.
// ---------------------------------------------------------------------------
__global__ void __launch_bounds__(256)
rope_update_kernel(const float* __restrict__ P,
                   const int* __restrict__ idx_ptr,
                   float* __restrict__ Qout,
                   float* __restrict__ out_cache)
{
    const int t = blockIdx.x * blockDim.x + threadIdx.x;
    const int idx = *idx_ptr;
    const float pos = (float)idx;

    if (t < 32768) {
        const int i = t & 127;
        const int n = (t >> 7) & 7;
        const int b = t >> 10;
        const float inv_ts = powf(10000.0f, -(float)i * (1.0f / 128.0f));
        const float r = pos * inv_ts;
        const float sn = sinf(r), cs = cosf(r);
        const float q1 = P[(size_t)b * PROJ_COLS + n * 256 + i];
        const float q2 = P[(size_t)b * PROJ_COLS + n * 256 + 128 + i];
        const size_t o = ((size_t)b * NQH + n) * HDIM + 2 * i;
        Qout[o]     = (q1 * cs - q2 * sn) * 0.0625f;   // 1/sqrt(256)
        Qout[o + 1] = (q2 * cs + q1 * sn) * 0.0625f;
    } else if (t < 32768 + 4096) {
        const int u = t - 32768;
        const int i = u & 127;
        const int b = u >> 7;
        const float inv_ts = powf(10000.0f, -(float)i * (1.0f / 128.0f));
        const float r = pos * inv_ts;
        const float sn = sinf(r), cs = cosf(r);
        const float k1 = P[(size_t)b * PROJ_COLS + 2048 + i];
        const float k2 = P[(size_t)b * PROJ_COLS + 2048 + 128 + i];
        const size_t o = (size_t)b * KV_BSTRIDE + (size_t)idx * HDIM + 2 * i;
        out_cache[o]     = k1 * cs - k2 * sn;
        out_cache[o + 1] = k2 * cs + k1 * sn;
    } else if (t < 45056) {
        const int u = t - 36864;
        const int h = u & 255;
        const int b = u >> 8;
        out_cache[(size_t)b * KV_BSTRIDE + (size_t)SEQ * HDIM +
                  (size_t)idx * HDIM + h] =
            P[(size_t)b * PROJ_COLS + 2304 + h];
    }
}

// ---------------------------------------------------------------------------
// K3: logits + fused K-cache copy.  One wave per 16 seq rows.
// 1024 blocks of 256 threads: block -> (b, 128-row chunk).
// Each lane holds q[n][lane*8 .. lane*8+7] for all 8 heads in registers.
// ---------------------------------------------------------------------------
__global__ void __launch_bounds__(256)
logits_kernel(const float* __restrict__ Q,
              const float* __restrict__ in_cache,
              float* __restrict__ out_cache,
              const unsigned char* __restrict__ mask,
              const int* __restrict__ idx_ptr,
              float* __restrict__ L)
{
    const int b = blockIdx.x >> 5;
    const int chunk = blockIdx.x & 31;
    const int wave = threadIdx.x >> 5;
    const int lane = threadIdx.x & 31;
    const int s_base = chunk * 128 + wave * 16;
    const int idx = *idx_ptr;

    // q fragments in registers: 8 heads x 8 floats per lane.
    float4 qa[NQH], qb[NQH];
    #pragma unroll
    for (int n = 0; n < NQH; ++n) {
        const float* qp = Q + ((size_t)b * NQH + n) * HDIM + lane * 8;
        qa[n] = *(const float4*)(qp);
        qb[n] = *(const float4*)(qp + 4);
    }

    for (int r = 0; r < 16; ++r) {
        const int s = s_base + r;
        const size_t off = (size_t)b * KV_BSTRIDE + (size_t)s * HDIM;
        const float* kp = ((s == idx) ? out_cache : in_cache) + off;
        if (s + 2 < SEQ) {
            __builtin_prefetch(in_cache + off + 2 * HDIM, 0, 1);
        }
        const float4 ka = ((const float4*)kp)[lane * 2];
        const float4 kb = ((const float4*)kp)[lane * 2 + 1];
        if (s != idx) {   // fused copy into the new cache
            float4* op = (float4*)(out_cache + off);
            op[lane * 2]     = ka;
            op[lane * 2 + 1] = kb;
        }
        const bool msk = mask[(size_t)b * SEQ + s] != 0;
        #pragma unroll
        for (int n = 0; n < NQH; ++n) {
            float p = ka.x * qa[n].x + ka.y * qa[n].y +
                      ka.z * qa[n].z + ka.w * qa[n].w +
                      kb.x * qb[n].x + kb.y * qb[n].y +
                      kb.z * qb[n].z + kb.w * qb[n].w;
            p += __shfl_xor(p, 16);
            p += __shfl_xor(p, 8);
            p += __shfl_xor(p, 4);
            p += __shfl_xor(p, 2);
            p += __shfl_xor(p, 1);
            if (lane == 0) {
                L[((size_t)b * NQH + n) * SEQ + s] = msk ? p : NEG_INF_F;
            }
        }
    }
}

// ---------------------------------------------------------------------------
// K4: softmax in place over L[b*8+n][4096]; zero probs if nothing attended.
// ---------------------------------------------------------------------------
__global__ void __launch_bounds__(256)
softmax_kernel(float* __restrict__ L)
{
    float* row = L + (size_t)blockIdx.x * SEQ;
    const int tid = threadIdx.x;
    __shared__ float red[256];

    float mx = -3.4e38f;
    for (int i = tid; i < SEQ; i += 256) mx = fmaxf(mx, row[i]);
    red[tid] = mx; __syncthreads();
    for (int o = 128; o > 0; o >>= 1) {
        if (tid < o) red[tid] = fmaxf(red[tid], red[tid + o]);
        __syncthreads();
    }
    const float m = red[0]; __syncthreads();

    float sum = 0.0f;
    for (int i = tid; i < SEQ; i += 256) {
        const float e = __expf(row[i] - m);
        row[i] = e;
        sum += e;
    }
    red[tid] = sum; __syncthreads();
    for (int o = 128; o > 0; o >>= 1) {
        if (tid < o) red[tid] += red[tid + o];
        __syncthreads();
    }
    // If every position was masked (max == NEG_INF) -> attn must be zero.
    const float inv = (m < -1.0e8f) ? 0.0f : (1.0f / red[0]);
    for (int i = tid; i < SEQ; i += 256) row[i] *= inv;
}

// ---------------------------------------------------------------------------
// K5: attn.V partials + fused V-cache copy.  512 blocks: (b, split of 256 s).
// Thread t owns column h = t; accumulates 8 heads locally (deterministic,
// no atomics).  Probs chunk staged in LDS — via the gfx1250 async
// global->LDS path (ASYNCcnt) when available.
// ---------------------------------------------------------------------------
__global__ void __launch_bounds__(256)
attnv_kernel(const float* __restrict__ L,
             const float* __restrict__ in_cache,
             float* __restrict__ out_cache,
             const int* __restrict__ idx_ptr,
             float* __restrict__ PT)
{
    const int b = blockIdx.x >> 4;
    const int split = blockIdx.x & 15;
    const int s0 = split * 256;
    const int h = threadIdx.x;
    const int idx = *idx_ptr;

    __shared__ float pl[NQH * 256];
#ifdef HAVE_ASYNC_LDS
    #pragma unroll
    for (int n = 0; n < NQH; ++n) {
        __builtin_amdgcn_global_load_async_to_lds_b32(
            (async_gptr*)&L[((size_t)b * NQH + n) * SEQ + s0 + h],
            (async_lptr*)&pl[n * 256 + h],
            0, 0);
    }
    __builtin_amdgcn_s_wait_asynccnt(0);
    __syncthreads();
#else
    #pragma unroll
    for (int n = 0; n < NQH; ++n) {
        pl[n * 256 + h] = L[((size_t)b * NQH + n) * SEQ + s0 + h];
    }
    __syncthreads();
#endif

    float acc[NQH];
    #pragma unroll
    for (int n = 0; n < NQH; ++n) acc[n] = 0.0f;

    const size_t vbase = (size_t)b * KV_BSTRIDE + (size_t)SEQ * HDIM;
    for (int sl = 0; sl < 256; ++sl) {
        const int s = s0 + sl;
        const size_t off = vbase + (size_t)s * HDIM + h;
        float v;
        if (s == idx) {
            v = out_cache[off];           // fresh V row written by K2
        } else {
            v = in_cache[off];
            out_cache[off] = v;           // fused copy into the new cache
            __builtin_prefetch(in_cache + off + HDIM, 0, 1);
        }
        #pragma unroll
        for (int n = 0; n < NQH; ++n) acc[n] += pl[n * 256 + sl] * v;
    }

    #pragma unroll
    for (int n = 0; n < NQH; ++n) {
        PT[((size_t)(b * 16 + split)) * 2048 + n * 256 + h] = acc[n];
    }
}

// ---------------------------------------------------------------------------
// K6: reduce split partials -> attn_flat[32][2048].
// ---------------------------------------------------------------------------
__global__ void __launch_bounds__(256)
reduce_kernel(const float* __restrict__ PT, float* __restrict__ A)
{
    const int e = blockIdx.x * 256 + threadIdx.x;   // 0..65535
    const int b = e >> 11;
    const int col = e & 2047;
    float s = 0.0f;
    #pragma unroll
    for (int split = 0; split < 16; ++split) {
        s += PT[((size_t)(b * 16 + split)) * 2048 + col];
    }
    A[e] = s;
}

// ---------------------------------------------------------------------------
// K7: output projection.  out[32][2048] = attn_flat[32][2048] @ Wo[2048][2048]
// (Wo (8,256,2048) flattens to row-major (2048,2048)).  WMMA f32, one wave
// per 16x16 tile: 2 * 128 = 256 waves.
// ---------------------------------------------------------------------------
__global__ void __launch_bounds__(256)
out_gemm_kernel(const float* __restrict__ A,
                const float* __restrict__ Wo,
                float* __restrict__ out)
{
    const int wave = (blockIdx.x * blockDim.x + threadIdx.x) >> 5;
    const int lane = threadIdx.x & 31;
    const int tile_m = wave / 128;          // 0..1
    const int tile_n = wave % 128;          // 0..127

    const int dcol = tile_n * 16 + (lane & 15);
    const float* wcol = Wo + dcol;          // stride DMODEL floats per k
    const int m = tile_m * 16 + (lane & 15);
    const int koff = (lane >> 4) << 1;

    v8f acc = {};
    #pragma unroll 2
    for (int k = 0; k < DMODEL; k += 4) {
        const int kk = k + koff;
        v2f a, bfrag;
        a.x = A[(size_t)m * DMODEL + kk];
        a.y = A[(size_t)m * DMODEL + kk + 1];
        bfrag.x = wcol[(size_t)kk * DMODEL];
        bfrag.y = wcol[(size_t)(kk + 1) * DMODEL];
        acc = __builtin_amdgcn_wmma_f32_16x16x4_f32(
            false, a, false, bfrag, (short)0, acc, false, false);
    }

    const int mrow_base = tile_m * 16 + ((lane >> 4) << 3);
    #pragma unroll
    for (int r = 0; r < 8; ++r) {
        out[(size_t)(mrow_base + r) * DMODEL + dcol] = acc[r];
    }
}

// ---------------------------------------------------------------------------
// Launcher
// ---------------------------------------------------------------------------
extern "C" void kernel_launch(void* const* d_in, const int* in_sizes, int n_in,
                              void* d_out, int out_size, void* d_ws, size_t ws_size,
                              hipStream_t stream) {
    const float* x            = (const float*)d_in[0];
    const unsigned char* mask = (const unsigned char*)d_in[1];
    const float* cache        = (const float*)d_in[2];
    const float* Wq           = (const float*)d_in[3];
    const float* Wk           = (const float*)d_in[4];
    const float* Wv           = (const float*)d_in[5];
    const float* Wo           = (const float*)d_in[6];
    const int*   idx          = (const int*)d_in[7];

    float* out_attn  = (float*)d_out;                    // [32][2048]
    float* out_cache = out_attn + BATCH * DMODEL;        // [32][2][4096][256]

    float* ws = (float*)d_ws;
    float* P  = ws;                    // [32][2560]        =   81920 f
    float* Q  = P  + BATCH * PROJ_COLS;// [32][8][256]      =   65536 f
    float* L  = Q  + BATCH * NQH * HDIM; // [32][8][4096]   = 1048576 f
    float* PT = L  + (size_t)BATCH * NQH * SEQ; // [32][16][2048] = 1048576 f
    float* A  = PT + (size_t)BATCH * 16 * 2048; // [32][2048]     =  65536 f

    qkv_gemm_kernel   <<<40,   256, 0, stream>>>(x, Wq, Wk, Wv, P);
    rope_update_kernel<<<176,  256, 0, stream>>>(P, idx, Q, out_cache);
    logits_kernel     <<<1024, 256, 0, stream>>>(Q, cache, out_cache, mask, idx, L);
    softmax_kernel    <<<256,  256, 0, stream>>>(L);
    attnv_kernel      <<<512,  256, 0, stream>>>(L, cache, out_cache, idx, PT);
    reduce_kernel     <<<256,  256, 0, stream>>>(PT, A);
    out_gemm_kernel   <<<32,   256, 0, stream>>>(A, Wo, out_attn);
}